// Istft_4423816314991
// MI455X (gfx1250) — compile-verified
//
#include <hip/hip_runtime.h>
#include <hip/hip_bf16.h>
#include <math.h>

// ---------------------------------------------------------------------------
// ISTFT on gfx1250: irfft2048 per frame via hermitian-pack -> 1024-pt complex
// IFFT (four-step 32x32) where every 32x32x32 real matmul runs on
// v_wmma_f32_16x16x4_f32 (full f32 precision on the matrix units).
// v2: DFT-32 matrix fragments come from a 32-entry LDS cos/sin table instead
// of per-fragment __sincosf (cuts the dominant VALU/trans cost per wave).
// ---------------------------------------------------------------------------

typedef __attribute__((ext_vector_type(2))) float v2f;
typedef __attribute__((ext_vector_type(8))) float v8f;

#define NBINS   1025
#define NFFT    2048
#define HOP     512
#define PI_F    3.14159265358979323846f

// D = A(16x4) * B(4x16) + C, f32 (neg flags must stay false for f32 WMMA)
static __device__ __forceinline__ v8f wmma_f32(v2f a, v2f b, v8f c) {
    return __builtin_amdgcn_wmma_f32_16x16x4_f32(
        false, a, false, b, (short)0, c, false, false);
}

__global__ void istft_zero_kernel(float* __restrict__ out, long long n) {
    long long i = (long long)blockIdx.x * blockDim.x + threadIdx.x;
    long long stride = (long long)gridDim.x * blockDim.x;
    for (; i < n; i += stride) out[i] = 0.0f;
}

__global__ __launch_bounds__(32)
void istft_frame_kernel(const float* __restrict__ sre,
                        const float* __restrict__ sim,
                        float* __restrict__ out,
                        int F, int out_len) {
    const int frame = blockIdx.x;            // 0 .. B*F-1
    const int b     = frame / F;
    const int f     = frame % F;
    const int lane  = threadIdx.x;           // wave32: 0..31
    const int lo16  = lane & 15;
    const int half  = lane >> 4;             // 0 for lanes 0-15, 1 for 16-31

    __shared__ float Xre[NBINS], Xim[NBINS];
    __shared__ float Zre[32 * 33], Zim[32 * 33];   // Zm[k1][k2], k = k1 + 32*k2
    __shared__ float Hre[32 * 33], Him[32 * 33];   // twiddled stage-1 result
    __shared__ float Fc[32], Fs[32], Fsn[32];      // DFT-32 table: cos/sin/-sin

    // one-time per wave: e^{+2*pi*i*p/32} table (p = 0..31)
    {
        float s, c;
        __sincosf((float)lane * (PI_F / 16.0f), &s, &c);
        Fc[lane]  = c;
        Fs[lane]  = s;
        Fsn[lane] = -s;
    }

    const float* xr = sre + (size_t)frame * NBINS;
    const float* xi = sim + (size_t)frame * NBINS;
    for (int k = lane; k < NBINS; k += 32) { Xre[k] = xr[k]; Xim[k] = xi[k]; }
    __syncthreads();

    // ---- Hermitian pack: Z[k] = 0.5*[(a+b) + i*e^{i*pi*k/1024}*(a-b)] ----
    for (int k = lane; k < 1024; k += 32) {
        float ar, ai, br, bi;
        if (k == 0) {                         // imag of bins 0 and 1024 ignored
            ar = Xre[0];     ai = 0.0f;
            br = Xre[1024];  bi = 0.0f;
        } else {
            ar = Xre[k];         ai = Xim[k];
            br = Xre[1024 - k];  bi = -Xim[1024 - k];   // conj
        }
        float sr = ar + br, si = ai + bi;     // a+b
        float dr = ar - br, di = ai - bi;     // a-b
        float s, c;
        __sincosf((float)k * (PI_F / 1024.0f), &s, &c);
        // i*(c+is)*(dr+i*di) = -(c*di + s*dr) + i*(c*dr - s*di)
        float zr = 0.5f * (sr - (c * di + s * dr));
        float zi = 0.5f * (si + (c * dr - s * di));
        int k1 = k & 31, k2 = k >> 5;
        Zre[k1 * 33 + k2] = zr;
        Zim[k1 * 33 + k2] = zi;
    }
    __syncthreads();

    // ---- Stage 1: G = Zm * F32, then twiddle -> H (in LDS) ----
    // F32[a][b] = e^{+2*pi*i*a*b/32} via LDS table
    for (int mi = 0; mi < 2; ++mi) {
        for (int ni = 0; ni < 2; ++ni) {
            v8f gre = {0.f,0.f,0.f,0.f,0.f,0.f,0.f,0.f};
            v8f gim = {0.f,0.f,0.f,0.f,0.f,0.f,0.f,0.f};
            const int mrow = mi * 16 + lo16;          // k1 row of A
            const int ncol = ni * 16 + lo16;          // n2 col of B
            for (int kk = 0; kk < 8; ++kk) {
                const int k0 = kk * 4 + 2 * half;     // K pair for this half-wave
                v2f are = { Zre[mrow * 33 + k0], Zre[mrow * 33 + k0 + 1] };
                v2f aim = { Zim[mrow * 33 + k0], Zim[mrow * 33 + k0 + 1] };
                const int p0 = (k0 * ncol) & 31;      // angle index for row k0
                const int p1 = (p0 + ncol) & 31;      // (k0+1)*ncol mod 32
                v2f brr = { Fc[p0],  Fc[p1]  };
                v2f bii = { Fs[p0],  Fs[p1]  };
                v2f bin = { Fsn[p0], Fsn[p1] };       // minus term
                gre = wmma_f32(are, brr, gre);
                gre = wmma_f32(aim, bin, gre);
                gim = wmma_f32(are, bii, gim);
                gim = wmma_f32(aim, brr, gim);
            }
            // twiddle e^{+2*pi*i*k1*n2/1024} and store to H
#pragma unroll
            for (int r = 0; r < 8; ++r) {
                int k1 = mi * 16 + r + 8 * half;
                int n2 = ni * 16 + lo16;
                float s, c;
                __sincosf((float)(k1 * n2) * (2.0f * PI_F / 1024.0f), &s, &c);
                float hr = gre[r] * c - gim[r] * s;
                float hi = gre[r] * s + gim[r] * c;
                Hre[k1 * 33 + n2] = hr;
                Him[k1 * 33 + n2] = hi;
            }
        }
    }
    __syncthreads();

    const float inv_n  = 1.0f / 1024.0f;
    const int   o_base = f * HOP - (NFFT / 2);        // trim = fft/2
    float* outb = out + (size_t)b * out_len;

    // ---- Stage 2: R = F32 * H ; unpack, window, overlap-add ----
    for (int pi = 0; pi < 2; ++pi) {
        for (int ni = 0; ni < 2; ++ni) {
            v8f rre = {0.f,0.f,0.f,0.f,0.f,0.f,0.f,0.f};
            v8f rim = {0.f,0.f,0.f,0.f,0.f,0.f,0.f,0.f};
            const int mrow = pi * 16 + lo16;          // n1 row of A = F32
            const int ncol = ni * 16 + lo16;          // n2 col of B = H
            for (int kk = 0; kk < 8; ++kk) {
                const int k0 = kk * 4 + 2 * half;
                const int p0 = (mrow * k0) & 31;      // angle index for col k0
                const int p1 = (p0 + mrow) & 31;      // mrow*(k0+1) mod 32
                v2f arr = { Fc[p0],  Fc[p1]  };
                v2f aii = { Fs[p0],  Fs[p1]  };
                v2f ain = { Fsn[p0], Fsn[p1] };
                v2f bre = { Hre[k0 * 33 + ncol], Hre[(k0 + 1) * 33 + ncol] };
                v2f bim = { Him[k0 * 33 + ncol], Him[(k0 + 1) * 33 + ncol] };
                rre = wmma_f32(arr, bre, rre);
                rre = wmma_f32(ain, bim, rre);
                rim = wmma_f32(arr, bim, rim);
                rim = wmma_f32(aii, bre, rim);
            }
#pragma unroll
            for (int r = 0; r < 8; ++r) {
                int n1 = pi * 16 + r + 8 * half;
                int n2 = ni * 16 + lo16;
                int n  = n2 + 32 * n1;                // time-pair index
                int t0 = 2 * n;
                int t1 = t0 + 1;
                float x0 = rre[r] * inv_n;            // x[2n]
                float x1 = rim[r] * inv_n;            // x[2n+1]
                float w0 = 0.5f - 0.5f * __cosf((float)t0 * (PI_F / 1024.0f));
                float w1 = 0.5f - 0.5f * __cosf((float)t1 * (PI_F / 1024.0f));
                int p0 = o_base + t0;
                int p1 = o_base + t1;
                if ((unsigned)p0 < (unsigned)out_len) atomicAdd(outb + p0, x0 * w0);
                if ((unsigned)p1 < (unsigned)out_len) atomicAdd(outb + p1, x1 * w1);
            }
        }
    }
}

__global__ void istft_norm_kernel(float* __restrict__ out, int B, int F, int out_len) {
    int j = blockIdx.x * blockDim.x + threadIdx.x;
    if (j >= out_len) return;
    int n   = j + (NFFT / 2);                         // untrimmed position
    int flo = (n - (NFFT - 1) + (HOP - 1)) >> 9;      // ceil((n-2047)/512)
    if (flo < 0) flo = 0;
    int fhi = n >> 9;
    if (fhi > F - 1) fhi = F - 1;
    float wsq = 0.0f;
    for (int fr = flo; fr <= fhi; ++fr) {
        int t = n - (fr << 9);
        float w = 0.5f - 0.5f * __cosf((float)t * (PI_F / 1024.0f));
        wsq += w * w;
    }
    float denom = (wsq > 1.17549435e-38f) ? wsq : 1.0f;
    float inv = 1.0f / denom;
    for (int b = 0; b < B; ++b)
        out[(size_t)b * out_len + j] *= inv;
}

extern "C" void kernel_launch(void* const* d_in, const int* in_sizes, int n_in,
                              void* d_out, int out_size, void* d_ws, size_t ws_size,
                              hipStream_t stream) {
    (void)n_in; (void)d_ws; (void)ws_size;
    const float* sre = (const float*)d_in[0];
    const float* sim = (const float*)d_in[1];
    float* out = (float*)d_out;

    // B*F = elems/NBINS ; out_size = B*(F-1)*HOP  =>  B = BF - out_size/HOP
    const int BF = in_sizes[0] / NBINS;               // 16000
    int B = BF - out_size / HOP;                      // 4
    if (B <= 0) B = 4;
    const int F = BF / B;                             // 4000
    const int out_len = (F - 1) * HOP;                // per-batch trimmed length

    // 1) zero accumulator (d_out doubles as the OLA accumulator)
    {
        long long n = (long long)B * out_len;
        int blocks = (int)((n + 255) / 256);
        if (blocks > 16384) blocks = 16384;
        istft_zero_kernel<<<blocks, 256, 0, stream>>>(out, n);
    }
    // 2) per-frame irfft (WMMA four-step) + window + overlap-add
    istft_frame_kernel<<<BF, 32, 0, stream>>>(sre, sim, out, F, out_len);
    // 3) divide by window_sumsquare (tiny-guarded), all batches
    istft_norm_kernel<<<(out_len + 255) / 256, 256, 0, stream>>>(out, B, F, out_len);
}